// SlayerModel_59236188946826
// MI455X (gfx1250) — compile-verified
//
#include <hip/hip_runtime.h>

// SLAYER spiking net forward on gfx1250.
// Pipeline: pool4 -> conv7(2->4, WMMA f32 16x16x4) -> spike -> pool4 ->
//           conv7(4->8, WMMA) -> spike -> pool4 -> dense(128->2).
// Memory-bound: 419MB input read once (~18us @ 23.3TB/s); all intermediates
// fit in the 192MB L2. fp32 WMMA (K=4 chunks) keeps exact f32 math.

#define T_DIM 100

typedef __attribute__((ext_vector_type(2))) float f32x2;
typedef __attribute__((ext_vector_type(8))) float f32x8;

__device__ __forceinline__ int clampi(int v, int lo, int hi) {
  return v < lo ? lo : (v > hi ? hi : v);
}

// ---------------- 4x4 average pool over [8,C,H,W,T] -> [8,C,H/4,W/4,T] ----
__global__ __launch_bounds__(256)
void avgpool4_k(const float* __restrict__ in, float* __restrict__ out,
                int C, int H, int W) {
  const int Ho = H >> 2, Wo = W >> 2;
  int idx = blockIdx.x * blockDim.x + threadIdx.x;
  int total = 8 * C * Ho * Wo;
  if (idx >= total) return;
  int wo = idx % Wo; int r = idx / Wo;
  int ho = r % Ho; r /= Ho;
  int c = r % C; int n = r / C;
  const float* base =
      in + (((size_t)(n * C + c) * H + ho * 4) * W + wo * 4) * T_DIM;
  float* ob = out + (size_t)idx * T_DIM;  // idx already == NCHoWo linear index
  const float s = 1.0f / 16.0f;
  for (int t = 0; t < T_DIM; t += 4) {  // T_DIM % 4 == 0, rows 400B -> aligned
    float4 acc = make_float4(0.f, 0.f, 0.f, 0.f);
#pragma unroll
    for (int dh = 0; dh < 4; ++dh) {
#pragma unroll
      for (int dw = 0; dw < 4; ++dw) {
        float4 v = *(const float4*)(base + ((size_t)dh * W + dw) * T_DIM + t);
        acc.x += v.x; acc.y += v.y; acc.z += v.z; acc.w += v.w;
      }
    }
    acc.x *= s; acc.y *= s; acc.z *= s; acc.w *= s;
    *(float4*)(ob + t) = acc;
  }
}

// ---------------- conv 7x7 pad 3 (per-timestep) as implicit GEMM via WMMA ---
// One wave32 computes one (n,h,w) pixel x 16 timesteps x all CO channels.
//   M = 16 timesteps (contiguous in memory), N = 16 (CO padded),
//   K = CI*7*8 (each 7-tap row padded to 8 so rows align to two K=4 chunks;
//   the pad tap is zero in BOTH A and B, so the contraction is unchanged).
// A 16x4 layout (ISA): lane L -> M=L%16; vgpr j -> K = 2*(L/16)+j.
// B 4x16 layout (row striped per VGPR, mirroring C/D): vgpr j, lane L ->
//   row = j + 2*(L/16), col = L%16.
// D: vgpr r, lane L -> M = r + 8*(L/16), N = L%16.
// All gather loads are UNCONDITIONAL with indices clamped into range; invalid
// taps are zeroed with a select afterwards. This keeps EXEC all-1s through the
// K-loop (WMMA requirement) and avoids per-load s_and_saveexec regions.
template <int CI, int CO, int HW>
__global__ __launch_bounds__(256)
void conv7_wmma_k(const float* __restrict__ in,   // [8,CI,HW,HW,T]
                  const float* __restrict__ wgt,  // [CO,CI,7,7]
                  float* __restrict__ out) {      // [8,CO,HW,HW,T]
  const int NTT = (T_DIM + 15) / 16;  // 7 time tiles
  const int gw = (int)((blockIdx.x * blockDim.x + threadIdx.x) >> 5);
  const int lane = threadIdx.x & 31;
  const int total = 8 * HW * HW * NTT;
  if (gw >= total) return;  // wave-uniform (grid is exact anyway)

  int tt = gw % NTT; int r = gw / NTT;
  int w = r % HW; r /= HW;
  int h = r % HW; int n = r / HW;

  const int t0 = tt * 16;
  const int m = lane & 15;     // A row (t offset) / B,D column (co)
  const int half = lane >> 4;  // lane half selects K pair / M upper half
  const int t = t0 + m;
  const bool tv = (t < T_DIM);
  const int tc = tv ? t : (T_DIM - 1);     // clamped (safe) t
  const int mc = (m < CO) ? m : 0;         // clamped (safe) co for B loads
  const bool mv = (m < CO);

  const float* inb = in + (size_t)n * CI * HW * HW * T_DIM;
  f32x8 acc = {0.f, 0.f, 0.f, 0.f, 0.f, 0.f, 0.f, 0.f};

  for (int ci = 0; ci < CI; ++ci) {
    const float* cplane = inb + (size_t)ci * HW * HW * T_DIM + tc;
    const float* wplane = wgt + (mc * CI + ci) * 49;
#pragma unroll
    for (int kh = 0; kh < 7; ++kh) {
      const int ih = h + kh - 3;
      const bool rowv = tv && (ih >= 0) && (ih < HW);
      const int ihc = clampi(ih, 0, HW - 1);
      const float* rowp = cplane + (size_t)ihc * HW * T_DIM;
      const float* wrow = wplane + kh * 7;
#pragma unroll
      for (int c2 = 0; c2 < 2; ++c2) {  // two K=4 chunks per padded row of 8
        f32x2 a, b;
#pragma unroll
        for (int j = 0; j < 2; ++j) {
          const int kw = c2 * 4 + 2 * half + j;  // 0..7; kw==7 is the pad tap
          const bool kv = (kw < 7);
          const int kwc = kv ? kw : 6;
          const int iw = w + kw - 3;
          const bool av_ok = rowv && kv && (iw >= 0) && (iw < HW);
          const int iwc = clampi(iw, 0, HW - 1);
          float bv = wrow[kwc];                     // unconditional, clamped
          float av = rowp[(size_t)iwc * T_DIM];     // unconditional, clamped
          a[j] = av_ok ? av : 0.f;
          b[j] = (kv && mv) ? bv : 0.f;
        }
        // D = A(16x4) * B(4x16) + C ; exact fp32
        acc = __builtin_amdgcn_wmma_f32_16x16x4_f32(false, a, false, b,
                                                    (short)0, acc, false, false);
      }
    }
  }

  const int co = m;
  if (co < CO) {
    float* ob = out + ((((size_t)n * CO + co) * HW + h) * HW + w) * T_DIM;
    const int tb = t0 + 8 * half;
#pragma unroll
    for (int rr = 0; rr < 8; ++rr)
      if (tb + rr < T_DIM) ob[tb + rr] = acc[rr];
  }
}

// ---------------- psp (cumsum over t) + IAF spike, in place ----------------
// v[t] = cumsum(x)[t] - count; s = (v >= 1); count += s   (theta = 1)
__global__ __launch_bounds__(256)
void spike_k(float* __restrict__ x, int total) {
  int idx = blockIdx.x * blockDim.x + threadIdx.x;
  if (idx >= total) return;
  float* p = x + (size_t)idx * T_DIM;
  float psp = 0.f, count = 0.f;
  for (int t = 0; t < T_DIM; t += 4) {
    float4 v4 = *(const float4*)(p + t);
    float e[4] = {v4.x, v4.y, v4.z, v4.w};
#pragma unroll
    for (int j = 0; j < 4; ++j) {
      psp += e[j];
      float vm = psp - count;
      float s = (vm >= 1.0f) ? 1.0f : 0.0f;
      count += s;
      e[j] = s;
    }
    *(float4*)(p + t) = make_float4(e[0], e[1], e[2], e[3]);
  }
}

// ---------------- dense: out[n,o,t] = sum_{c,h,w} p2[n,c,h,w,t]*wd[o,c,h,w] -
__global__ __launch_bounds__(256)
void dense_k(const float* __restrict__ p2, const float* __restrict__ wd,
             float* __restrict__ out) {
  int idx = blockIdx.x * blockDim.x + threadIdx.x;
  if (idx >= 8 * 2 * T_DIM) return;
  int t = idx % T_DIM; int r = idx / T_DIM;
  int o = r & 1; int n = r >> 1;
  float acc = 0.f;
  for (int c = 0; c < 8; ++c)
#pragma unroll
    for (int h = 0; h < 4; ++h)
#pragma unroll
      for (int w = 0; w < 4; ++w)
        acc += p2[((((size_t)n * 8 + c) * 4 + h) * 4 + w) * T_DIM + t] *
               wd[((o * 8 + c) * 4 + h) * 4 + w];
  out[(n * 2 + o) * T_DIM + t] = acc;
}

extern "C" void kernel_launch(void* const* d_in, const int* in_sizes, int n_in,
                              void* d_out, int out_size, void* d_ws,
                              size_t ws_size, hipStream_t stream) {
  const float* data = (const float*)d_in[0];  // [8,2,256,256,100]
  const float* w0   = (const float*)d_in[1];  // [4,2,7,7,1]
  const float* w1   = (const float*)d_in[2];  // [8,4,7,7,1]
  const float* wd   = (const float*)d_in[3];  // [2,8,4,4,1]
  float* outp = (float*)d_out;                // [8,2,100]

  float* ws = (float*)d_ws;
  float* P0 = ws;                                   // [8,2,64,64,100] 26.2MB
  float* X0 = P0 + (size_t)8 * 2 * 64 * 64 * T_DIM; // [8,4,64,64,100] 52.4MB (conv0 out, spiked in place)
  float* P1 = X0 + (size_t)8 * 4 * 64 * 64 * T_DIM; // [8,4,16,16,100] 3.3MB
  float* X1 = P1 + (size_t)8 * 4 * 16 * 16 * T_DIM; // [8,8,16,16,100] 6.6MB (conv1 out, spiked in place)
  float* P2 = X1 + (size_t)8 * 8 * 16 * 16 * T_DIM; // [8,8,4,4,100]   0.4MB

  // Stage 0: pool input (the only HBM-heavy pass; 419MB read once)
  avgpool4_k<<<(8 * 2 * 64 * 64 + 255) / 256, 256, 0, stream>>>(
      data, P0, 2, 256, 256);

  // Stage 1: conv0 via WMMA; 8*64*64*7 waves, 8 waves/block
  conv7_wmma_k<2, 4, 64><<<(8 * 64 * 64 * 7) / 8, 256, 0, stream>>>(P0, w0, X0);
  spike_k<<<(8 * 4 * 64 * 64 + 255) / 256, 256, 0, stream>>>(X0, 8 * 4 * 64 * 64);

  // Stage 2
  avgpool4_k<<<(8 * 4 * 16 * 16 + 255) / 256, 256, 0, stream>>>(
      X0, P1, 4, 64, 64);
  conv7_wmma_k<4, 8, 16><<<(8 * 16 * 16 * 7) / 8, 256, 0, stream>>>(P1, w1, X1);
  spike_k<<<(8 * 8 * 16 * 16 + 255) / 256, 256, 0, stream>>>(X1, 8 * 8 * 16 * 16);

  // Stage 3
  avgpool4_k<<<(8 * 8 * 4 * 4 + 255) / 256, 256, 0, stream>>>(X1, P2, 8, 16, 16);
  dense_k<<<(8 * 2 * T_DIM + 255) / 256, 256, 0, stream>>>(P2, wd, outp);
}